// GNNEncoder_317827579954
// MI455X (gfx1250) — compile-verified
//
#include <hip/hip_runtime.h>
#include <hip/hip_bf16.h>

// ---------------------------------------------------------------------------
// GCN 2-layer forward for MI455X (gfx1250, wave32).
//   h1 = relu( Agg( x@W1 ) + b1 );  out = Agg( h1@W2 ) + b2
// where Agg scatters  dinv[src]*dinv[dst] * row[src]  into row[dst]
// over 1.6M edges + 50K self loops (dst-side degree, rsqrt normalization).
//
// GEMMs use V_WMMA_F32_16X16X4_F32 (fp32-exact matrix pipe).
// Aggregation: one wave per edge, 8 strided f32 atomics per lane into L2.
// ---------------------------------------------------------------------------

typedef __attribute__((ext_vector_type(2))) float v2f;
typedef __attribute__((ext_vector_type(8))) float v8f;

#define HID 256   // in_ch == hid == out_ch == 256

// ---------------- degree / normalization --------------------------------
__global__ __launch_bounds__(256) void k_deg_init(float* __restrict__ deg, int n) {
    int i = blockIdx.x * 256 + threadIdx.x;
    if (i < n) deg[i] = 1.0f;                  // self-loop contributes 1
}

__global__ __launch_bounds__(256) void k_deg_count(const int* __restrict__ dst,
                                                   float* __restrict__ deg, int n_edges) {
    int e = blockIdx.x * 256 + threadIdx.x;
    if (e < n_edges) atomicAdd(&deg[dst[e]], 1.0f);
}

__global__ __launch_bounds__(256) void k_dinv(float* __restrict__ deg, int n) {
    int i = blockIdx.x * 256 + threadIdx.x;
    if (i < n) {
        float d = deg[i];
        deg[i] = (d > 0.0f) ? __frsqrt_rn(d) : 0.0f;   // v_rsq_f32
    }
}

// ---------------- fp32 WMMA GEMM:  C[n_rows x 256] = A[n_rows x 256] @ W[256 x 256]
// Block = 128 threads = 4 waves. blockIdx.x selects a 16-row strip.
// Wave w computes cols [w*64, w*64+64) as four 16x16 WMMA accumulators,
// reusing one A fragment per K-step across the 4 column tiles.
__global__ __launch_bounds__(128) void k_gemm_wmma_f32(const float* __restrict__ A,
                                                       const float* __restrict__ W,
                                                       float* __restrict__ C,
                                                       int n_rows) {
    const int lane = threadIdx.x & 31;
    const int wave = threadIdx.x >> 5;        // 0..3
    const int row0 = blockIdx.x * 16;
    const int col0 = wave * 64;
    const int half = lane >> 4;               // 0: lanes 0-15, 1: lanes 16-31
    const int l16  = lane & 15;
    if (row0 >= n_rows) return;

    v8f acc0 = {}, acc1 = {}, acc2 = {}, acc3 = {};

    for (int k = 0; k < HID; k += 4) {
        // A fragment (16x4 f32): lane half 0 holds K=k,k+1; half 1 holds K=k+2,k+3
        const float* ap = A + (size_t)(row0 + l16) * HID + (k + half * 2);
        v2f afrag = *(const v2f*)ap;

        // B fragments (4x16 f32): VGPR j, half h -> W[k + h*2 + j][col0 + t*16 + l16]
        const float* w0 = W + (size_t)(k + half * 2) * HID + col0 + l16;
        const float* w1 = w0 + HID;
        v2f b0, b1, b2, b3;
        b0.x = w0[0];   b0.y = w1[0];
        b1.x = w0[16];  b1.y = w1[16];
        b2.x = w0[32];  b2.y = w1[32];
        b3.x = w0[48];  b3.y = w1[48];

        acc0 = __builtin_amdgcn_wmma_f32_16x16x4_f32(false, afrag, false, b0,
                                                     (short)0, acc0, false, false);
        acc1 = __builtin_amdgcn_wmma_f32_16x16x4_f32(false, afrag, false, b1,
                                                     (short)0, acc1, false, false);
        acc2 = __builtin_amdgcn_wmma_f32_16x16x4_f32(false, afrag, false, b2,
                                                     (short)0, acc2, false, false);
        acc3 = __builtin_amdgcn_wmma_f32_16x16x4_f32(false, afrag, false, b3,
                                                     (short)0, acc3, false, false);
    }

    // C/D layout: VGPR v, half h -> row (v + 8h), col = l16 (+ tile offset)
    #pragma unroll
    for (int v = 0; v < 8; ++v) {
        float* cp = C + (size_t)(row0 + v + half * 8) * HID + col0 + l16;
        cp[0]  = acc0[v];
        cp[16] = acc1[v];
        cp[32] = acc2[v];
        cp[48] = acc3[v];
    }
}

// ---------------- bias broadcast init -----------------------------------
__global__ __launch_bounds__(256) void k_bias_init(float* __restrict__ out,
                                                   const float* __restrict__ b,
                                                   size_t total) {
    size_t i = (size_t)blockIdx.x * 256 + threadIdx.x;
    if (i < total) out[i] = b[i & (HID - 1)];
}

// ---------------- edge scatter-aggregate --------------------------------
// One wave per edge (8 edges per 256-thread block). Lane handles channels
// lane, lane+32, ..., lane+224 -> fully coalesced gathers, coalesced atomics.
__global__ __launch_bounds__(256) void k_scatter(const float* __restrict__ h,
                                                 const int* __restrict__ src,
                                                 const int* __restrict__ dst,
                                                 const float* __restrict__ dinv,
                                                 float* __restrict__ out,
                                                 int n_edges, int n_nodes) {
    int e = blockIdx.x * 8 + (threadIdx.x >> 5);
    int lane = threadIdx.x & 31;
    int total = n_edges + n_nodes;
    if (e >= total) return;

    int s, d;
    if (e < n_edges) { s = src[e]; d = dst[e]; }
    else             { s = e - n_edges; d = s; }   // self loop

    float nrm = dinv[s] * dinv[d];
    const float* hp = h + (size_t)s * HID;
    float* op = out + (size_t)d * HID;

    #pragma unroll
    for (int j = 0; j < 8; ++j) {
        int c = lane + j * 32;
        atomicAdd(op + c, nrm * hp[c]);
    }
}

// ---------------- relu ---------------------------------------------------
__global__ __launch_bounds__(256) void k_relu(float* __restrict__ x, size_t total) {
    size_t i = (size_t)blockIdx.x * 256 + threadIdx.x;
    if (i < total) x[i] = fmaxf(x[i], 0.0f);
}

// ---------------------------------------------------------------------------
extern "C" void kernel_launch(void* const* d_in, const int* in_sizes, int n_in,
                              void* d_out, int out_size, void* d_ws, size_t ws_size,
                              hipStream_t stream) {
    const float* x   = (const float*)d_in[0];           // [N, 256]
    const int*   ei  = (const int*)d_in[1];             // [2, E] (row-major)
    const float* W1  = (const float*)d_in[2];           // [256, 256]
    const float* b1  = (const float*)d_in[3];           // [256]
    const float* W2  = (const float*)d_in[4];           // [256, 256]
    const float* b2  = (const float*)d_in[5];           // [256]
    float*       out = (float*)d_out;                   // [N, 256]

    const int n_nodes = in_sizes[0] / HID;              // 50000
    const int n_edges = in_sizes[1] / 2;                // 1600000
    const int* srcp = ei;
    const int* dstp = ei + n_edges;

    // workspace: dinv[N] | t1[N*256] | t2[N*256]
    float* dinv = (float*)d_ws;
    float* t1   = dinv + n_nodes;
    float* t2   = t1 + (size_t)n_nodes * HID;

    const size_t feat_total = (size_t)n_nodes * HID;
    const int nb_nodes = (n_nodes + 255) / 256;
    const int nb_edges = (n_edges + 255) / 256;
    const int nb_feat  = (int)((feat_total + 255) / 256);
    const int nb_rows  = (n_nodes + 15) / 16;                  // 16-row GEMM strips
    const int nb_scat  = (n_edges + n_nodes + 7) / 8;          // 8 edges per block

    // degree -> dinv
    k_deg_init <<<nb_nodes, 256, 0, stream>>>(dinv, n_nodes);
    k_deg_count<<<nb_edges, 256, 0, stream>>>(dstp, dinv, n_edges);
    k_dinv     <<<nb_nodes, 256, 0, stream>>>(dinv, n_nodes);

    // layer 1: t1 = x@W1 ; t2 = b1 + Agg(t1) ; relu(t2)
    k_gemm_wmma_f32<<<nb_rows, 128, 0, stream>>>(x, W1, t1, n_nodes);
    k_bias_init    <<<nb_feat, 256, 0, stream>>>(t2, b1, feat_total);
    k_scatter      <<<nb_scat, 256, 0, stream>>>(t1, srcp, dstp, dinv, t2,
                                                 n_edges, n_nodes);
    k_relu         <<<nb_feat, 256, 0, stream>>>(t2, feat_total);

    // layer 2: t1 = t2@W2 ; out = b2 + Agg(t1)
    k_gemm_wmma_f32<<<nb_rows, 128, 0, stream>>>(t2, W2, t1, n_nodes);
    k_bias_init    <<<nb_feat, 256, 0, stream>>>(out, b2, feat_total);
    k_scatter      <<<nb_scat, 256, 0, stream>>>(t1, srcp, dstp, dinv, out,
                                                 n_edges, n_nodes);
}